// GNNClassifier_75093208203283
// MI455X (gfx1250) — compile-verified
//
#include <hip/hip_runtime.h>
#include <hip/hip_bf16.h>

typedef __attribute__((ext_vector_type(2))) float v2f;
typedef __attribute__((ext_vector_type(8))) float v8f;

#define C_IN   128
#define HDIM   128
#define NCLS   10
#define NGRAPH 128

// ---------------------------------------------------------------------------
// h = x @ W_conv  via V_WMMA_F32_16X16X4_F32.
// Block = 256 threads = 8 waves; block covers a 128-row slab of x.
// W (128x128 fp32, 64KB) staged in LDS, reused by all 8 waves.
// Wave w computes rows [blk*128 + w*16, +16): A-frags reused across the 8
// 16-wide column tiles (8 v8f accumulators live in registers).
// ---------------------------------------------------------------------------
__global__ void __launch_bounds__(256) gemm_xw_wmma(const float* __restrict__ x,
                                                    const float* __restrict__ W,
                                                    float* __restrict__ h,
                                                    int N) {
    __shared__ float sW[C_IN * HDIM];   // 64 KB
    const int t = threadIdx.x;
    #pragma unroll
    for (int i = 0; i < (C_IN * HDIM) / 256; ++i)
        sW[t + i * 256] = W[t + i * 256];
    __syncthreads();

    const int wave  = t >> 5;
    const int lane  = t & 31;
    const int mloc  = lane & 15;     // row within tile (A/C), col within tile (B/C)
    const int khalf = lane >> 4;     // 0: K pair {0,1} / rows 0..7 ; 1: K pair {2,3} / rows 8..15
    const int rowBase = blockIdx.x * 128 + wave * 16;
    if (rowBase >= N) return;        // uniform per wave -> EXEC stays all-1s for WMMA

    v8f acc[8];
    #pragma unroll
    for (int c = 0; c < 8; ++c) acc[c] = (v8f){};

    const float* xrow = x + (size_t)(rowBase + mloc) * C_IN;

    for (int k0 = 0; k0 < C_IN; k0 += 4) {
        // A 16x4 fp32 frag (ISA 7.12.2): VGPR0 = K0(lo lanes)/K2(hi), VGPR1 = K1/K3
        v2f a;
        a.x = xrow[k0 + 2 * khalf + 0];
        a.y = xrow[k0 + 2 * khalf + 1];
        #pragma unroll
        for (int c = 0; c < 8; ++c) {
            v2f b;   // B 4x16: row K striped across lanes
            b.x = sW[(k0 + 2 * khalf + 0) * HDIM + c * 16 + mloc];
            b.y = sW[(k0 + 2 * khalf + 1) * HDIM + c * 16 + mloc];
            acc[c] = __builtin_amdgcn_wmma_f32_16x16x4_f32(
                false, a, false, b, (short)0, acc[c], false, false);
        }
    }

    // C/D layout: VGPR r -> (M=r, N=lane) lanes 0-15, (M=r+8, N=lane-16) lanes 16-31
    #pragma unroll
    for (int c = 0; c < 8; ++c) {
        #pragma unroll
        for (int r = 0; r < 8; ++r) {
            int row = rowBase + r + 8 * khalf;
            h[(size_t)row * HDIM + c * 16 + mloc] = acc[c][r];
        }
    }
}

// ---------------------------------------------------------------------------
// Degree / normalization
// ---------------------------------------------------------------------------
__global__ void deg_init(float* __restrict__ deg, int N) {
    int i = blockIdx.x * blockDim.x + threadIdx.x;
    if (i < N) deg[i] = 1.0f;                 // self-loop contribution
}

__global__ void deg_accum(const int* __restrict__ col, float* __restrict__ deg, int E) {
    int i = blockIdx.x * blockDim.x + threadIdx.x;
    if (i < E) atomicAdd(&deg[col[i]], 1.0f);
}

__global__ void dinv_kernel(const float* __restrict__ deg, float* __restrict__ dinv, int N) {
    int i = blockIdx.x * blockDim.x + threadIdx.x;
    if (i < N) {
        float d = deg[i];
        dinv[i] = d > 0.0f ? rsqrtf(d) : 0.0f;
    }
}

// out[i,:] = h[i,:] * dinv[i]^2 (self-loop message) + b_conv  -- also zero-inits out
__global__ void init_out(const float* __restrict__ h, const float* __restrict__ dinv,
                         const float* __restrict__ b_conv, float* __restrict__ out,
                         long total) {
    long idx = (long)blockIdx.x * blockDim.x + threadIdx.x;
    if (idx < total) {
        int i = (int)(idx >> 7);
        int j = (int)(idx & 127);
        float d = dinv[i];
        out[idx] = h[idx] * d * d + b_conv[j];
    }
}

// ---------------------------------------------------------------------------
// Message passing: one wave per edge, each lane carries 4 channels (float4).
// ~820 MB gather + ~820 MB atomic scatter; h/out fit in the 192 MB L2, so
// global_atomic_add_f32 resolves near-chip.
// ---------------------------------------------------------------------------
__global__ void __launch_bounds__(256) edge_msgs(const int* __restrict__ row,
                                                 const int* __restrict__ col,
                                                 const float* __restrict__ h,
                                                 const float* __restrict__ dinv,
                                                 float* __restrict__ out, int E) {
    const int wavesPerBlock = blockDim.x >> 5;
    const int lane = threadIdx.x & 31;
    const int totalWaves = gridDim.x * wavesPerBlock;
    for (int e = blockIdx.x * wavesPerBlock + (threadIdx.x >> 5); e < E; e += totalWaves) {
        int r = row[e], c = col[e];
        float w = dinv[r] * dinv[c];
        float4 v = ((const float4*)(h + (size_t)r * HDIM))[lane];
        float* o = out + (size_t)c * HDIM + lane * 4;
        atomicAdd(o + 0, v.x * w);
        atomicAdd(o + 1, v.y * w);
        atomicAdd(o + 2, v.z * w);
        atomicAdd(o + 3, v.w * w);
    }
}

// ---------------------------------------------------------------------------
// Global mean pool
// ---------------------------------------------------------------------------
__global__ void zero_pool(float* __restrict__ pooled, float* __restrict__ counts) {
    int i = blockIdx.x * blockDim.x + threadIdx.x;
    if (i < NGRAPH * HDIM) pooled[i] = 0.0f;
    if (i < NGRAPH) counts[i] = 0.0f;
}

__global__ void __launch_bounds__(256) pool_kernel(const float* __restrict__ out,
                                                   const int* __restrict__ batch,
                                                   float* __restrict__ pooled,
                                                   float* __restrict__ counts, int N) {
    const int wavesPerBlock = blockDim.x >> 5;
    const int lane = threadIdx.x & 31;
    const int totalWaves = gridDim.x * wavesPerBlock;
    for (int i = blockIdx.x * wavesPerBlock + (threadIdx.x >> 5); i < N; i += totalWaves) {
        int g = batch[i];
        float4 v = ((const float4*)(out + (size_t)i * HDIM))[lane];
        float* p = pooled + g * HDIM + lane * 4;
        atomicAdd(p + 0, v.x);
        atomicAdd(p + 1, v.y);
        atomicAdd(p + 2, v.z);
        atomicAdd(p + 3, v.w);
        if (lane == 0) atomicAdd(&counts[g], 1.0f);
    }
}

// ---------------------------------------------------------------------------
// Head: (pooled / count) @ W_lin + b_lin  -> [128, 10]  (tiny)
// ---------------------------------------------------------------------------
__global__ void head_kernel(const float* __restrict__ pooled, const float* __restrict__ counts,
                            const float* __restrict__ W_lin, const float* __restrict__ b_lin,
                            float* __restrict__ outp) {
    int tid = blockIdx.x * blockDim.x + threadIdx.x;
    if (tid < NGRAPH * NCLS) {
        int g = tid / NCLS, c = tid % NCLS;
        float inv = 1.0f / fmaxf(counts[g], 1.0f);
        float s = 0.0f;
        #pragma unroll 4
        for (int k = 0; k < HDIM; ++k)
            s += pooled[g * HDIM + k] * W_lin[k * NCLS + c];
        outp[tid] = s * inv + b_lin[c];
    }
}

extern "C" void kernel_launch(void* const* d_in, const int* in_sizes, int n_in,
                              void* d_out, int out_size, void* d_ws, size_t ws_size,
                              hipStream_t stream) {
    const float* x      = (const float*)d_in[0];
    const int*   edges  = (const int*)  d_in[1];
    const int*   batch  = (const int*)  d_in[2];
    const float* W_conv = (const float*)d_in[3];
    const float* b_conv = (const float*)d_in[4];
    const float* W_lin  = (const float*)d_in[5];
    const float* b_lin  = (const float*)d_in[6];
    float*       outp   = (float*)d_out;

    const int N = in_sizes[0] / C_IN;
    const int E = in_sizes[1] / 2;
    const int* row = edges;       // edge_index[0] : source
    const int* col = edges + E;   // edge_index[1] : target

    // workspace layout (fp32): h[N*128] | out[N*128] | deg[N] | dinv[N] | pooled[128*128] | counts[128]
    float* ws     = (float*)d_ws;
    float* h      = ws;
    float* oconv  = h + (size_t)N * HDIM;
    float* deg    = oconv + (size_t)N * HDIM;
    float* dinv   = deg + N;
    float* pooled = dinv + N;
    float* counts = pooled + NGRAPH * HDIM;

    deg_init   <<<(N + 255) / 256, 256, 0, stream>>>(deg, N);
    deg_accum  <<<(E + 255) / 256, 256, 0, stream>>>(col, deg, E);
    dinv_kernel<<<(N + 255) / 256, 256, 0, stream>>>(deg, dinv, N);

    gemm_xw_wmma<<<(N + 127) / 128, 256, 0, stream>>>(x, W_conv, h, N);

    long total = (long)N * HDIM;
    init_out<<<(int)((total + 255) / 256), 256, 0, stream>>>(h, dinv, b_conv, oconv, total);

    edge_msgs<<<(E + 7) / 8, 256, 0, stream>>>(row, col, h, dinv, oconv, E);

    zero_pool  <<<(NGRAPH * HDIM + 255) / 256, 256, 0, stream>>>(pooled, counts);
    pool_kernel<<<(N + 7) / 8, 256, 0, stream>>>(oconv, batch, pooled, counts, N);

    head_kernel<<<(NGRAPH * NCLS + 255) / 256, 256, 0, stream>>>(pooled, counts, W_lin, b_lin, outp);
}